// FSMNKernel_82205674045742
// MI455X (gfx1250) — compile-verified
//
#include <hip/hip_runtime.h>
#include <cstdint>

// FSMN depthwise temporal FIR, fp32.
//   out[b,t,d] = sum_k filt[k,d] * x[b, t+off_k, d]
// off_k: left taps  k=0..19  -> 2k-40  (even, -40..-2)
//        center     k=20     -> 0
//        right taps k=21..40 -> 2k-41  (odd, +1..+39)
// Zero padding outside [0, T).

constexpr int B = 32;
constexpr int T = 2000;
constexpr int D = 512;
constexpr int L_ORDER = 20;
constexpr int NTAP    = 41;     // L + R + 1
constexpr int HALO    = 40;     // L_ORDER*L_STRIDE == R_ORDER*R_STRIDE

constexpr int T_TILE  = 128;    // 16 tiles cover T=2000 (last tile partial)
constexpr int D_TILE  = 64;     // 8 chunks cover D=512 exactly
constexpr int W       = T_TILE + 2 * HALO;   // 208 window rows in LDS
constexpr int R_PER_T = T_TILE / 16;         // 8 outputs per thread
constexpr int N_TTILE = (T + T_TILE - 1) / T_TILE;  // 16

#define LD4(p) (*(const float4*)(p))

__global__ __launch_bounds__(256)
void fsmn_fir_kernel(const float* __restrict__ x,
                     const float* __restrict__ filt,
                     float* __restrict__ out) {
    __shared__ float xs[W][D_TILE];       // 52 KB input window (incl. halo)
    __shared__ float fs[NTAP][D_TILE];    // 10.25 KB filter chunk

    const int tid = threadIdx.x;
    const int t0  = blockIdx.x * T_TILE;
    const int d0  = blockIdx.y * D_TILE;
    const int b   = blockIdx.z;

    const float* xb = x + (size_t)b * T * D;

    // ---------------- load phase ----------------
    const int row0 = tid >> 4;          // 0..15
    const int dl   = (tid & 15) * 4;    // float4 column within the 64-ch chunk
    #pragma unroll
    for (int p = 0; p < W / 16; ++p) {
        const int row = p * 16 + row0;
        const int t   = t0 - HALO + row;
        const uint32_t lds = (uint32_t)(uintptr_t)(&xs[row][dl]);
        if (t >= 0 && t < T) {
            // CDNA5 async copy: global -> LDS, 16B per lane, no VGPR staging.
            const uint64_t ga =
                (uint64_t)(uintptr_t)(xb + (size_t)t * D + d0 + dl);
            asm volatile("global_load_async_to_lds_b128 %0, %1, off"
                         :: "v"(lds), "v"(ga) : "memory");
        } else {
            xs[row][dl + 0] = 0.0f;
            xs[row][dl + 1] = 0.0f;
            xs[row][dl + 2] = 0.0f;
            xs[row][dl + 3] = 0.0f;
        }
    }

    // Filter chunk: 41 taps x 16 float4 columns.
    for (int idx = tid; idx < NTAP * 16; idx += 256) {
        const int k = idx >> 4;
        const int c = (idx & 15) * 4;
        *(float4*)(&fs[k][c]) = LD4(filt + (size_t)k * D + d0 + c);
    }

    asm volatile("s_wait_asynccnt 0x0" ::: "memory");
    __syncthreads();

    // ---------------- compute phase ----------------
    const int tg = tid >> 4;             // 0..15 : t-group
    const int tl = tg * R_PER_T;         // local output t base (0..120)

    float4 acc[R_PER_T];
    #pragma unroll
    for (int r = 0; r < R_PER_T; ++r)
        acc[r] = make_float4(0.0f, 0.0f, 0.0f, 0.0f);

    // Rotating 8-row register window, slot = (row index rel. phase base) & 7.
    // Tap groups of 4 keep all slot indices compile-time constant while the
    // GROUP loops stay rolled -> bounded register lifetimes, no VGPR-MSB use.
    float4 w[8];

    // ===== even-offset taps: k=0..19 in 5 groups of 4, then center k=20 =====
    #pragma unroll
    for (int s = 0; s < 8; ++s)
        w[s] = LD4(&xs[tl + s][dl]);

    #pragma clang loop unroll(disable)
    for (int g = 0; g < 5; ++g) {
        #pragma unroll
        for (int i = 0; i < 4; ++i) {
            const float4 f = LD4(&fs[4 * g + i][dl]);
            #pragma unroll
            for (int r = 0; r < R_PER_T; ++r) {
                const float4 v = w[(2 * i + r) & 7];
                acc[r].x = fmaf(v.x, f.x, acc[r].x);
                acc[r].y = fmaf(v.y, f.y, acc[r].y);
                acc[r].z = fmaf(v.z, f.z, acc[r].z);
                acc[r].w = fmaf(v.w, f.w, acc[r].w);
            }
            const int n = 8 * g + 2 * i + 8;   // next two window rows
            w[(2 * i) & 7]     = LD4(&xs[tl + n][dl]);
            w[(2 * i + 1) & 7] = LD4(&xs[tl + n + 1][dl]);
        }
    }
    {   // center tap k=20: window now holds rows tl+40 .. tl+47, slot r = row r.
        const float4 f = LD4(&fs[L_ORDER][dl]);
        #pragma unroll
        for (int r = 0; r < R_PER_T; ++r) {
            const float4 v = w[r];
            acc[r].x = fmaf(v.x, f.x, acc[r].x);
            acc[r].y = fmaf(v.y, f.y, acc[r].y);
            acc[r].z = fmaf(v.z, f.z, acc[r].z);
            acc[r].w = fmaf(v.w, f.w, acc[r].w);
        }
    }

    // ===== odd-offset taps: j=0..19 (filter 21+j), base row tl+41 =====
    #pragma unroll
    for (int s = 0; s < 8; ++s)
        w[s] = LD4(&xs[tl + HALO + 1 + s][dl]);

    #pragma clang loop unroll(disable)
    for (int g = 0; g < 4; ++g) {                  // taps j = 0..15
        #pragma unroll
        for (int i = 0; i < 4; ++i) {
            const float4 f = LD4(&fs[L_ORDER + 1 + 4 * g + i][dl]);
            #pragma unroll
            for (int r = 0; r < R_PER_T; ++r) {
                const float4 v = w[(2 * i + r) & 7];
                acc[r].x = fmaf(v.x, f.x, acc[r].x);
                acc[r].y = fmaf(v.y, f.y, acc[r].y);
                acc[r].z = fmaf(v.z, f.z, acc[r].z);
                acc[r].w = fmaf(v.w, f.w, acc[r].w);
            }
            const int n = 8 * g + 2 * i + 8;
            w[(2 * i) & 7]     = LD4(&xs[tl + HALO + 1 + n][dl]);
            w[(2 * i + 1) & 7] = LD4(&xs[tl + HALO + 1 + n + 1][dl]);
        }
    }
    #pragma unroll
    for (int i = 0; i < 4; ++i) {                  // peeled taps j = 16..19
        const float4 f = LD4(&fs[L_ORDER + 17 + i][dl]);
        #pragma unroll
        for (int r = 0; r < R_PER_T; ++r) {
            const float4 v = w[(2 * i + r) & 7];
            acc[r].x = fmaf(v.x, f.x, acc[r].x);
            acc[r].y = fmaf(v.y, f.y, acc[r].y);
            acc[r].z = fmaf(v.z, f.z, acc[r].z);
            acc[r].w = fmaf(v.w, f.w, acc[r].w);
        }
        if (i < 3) {                               // no loads past row tl+86
            const int n = 32 + 2 * i + 8;
            w[(2 * i) & 7]     = LD4(&xs[tl + HALO + 1 + n][dl]);
            w[(2 * i + 1) & 7] = LD4(&xs[tl + HALO + 1 + n + 1][dl]);
        }
    }

    // ---------------- store phase ----------------
    float* ob = out + (size_t)b * T * D + (size_t)t0 * D + d0;
    #pragma unroll
    for (int r = 0; r < R_PER_T; ++r) {
        if (t0 + tl + r < T) {
            *(float4*)(ob + (size_t)(tl + r) * D + dl) = acc[r];
        }
    }
}

extern "C" void kernel_launch(void* const* d_in, const int* in_sizes, int n_in,
                              void* d_out, int out_size, void* d_ws, size_t ws_size,
                              hipStream_t stream) {
    const float* x    = (const float*)d_in[0];   // [B, T, D] fp32
    const float* filt = (const float*)d_in[1];   // [41, D]   fp32
    float* out        = (float*)d_out;           // [B, T, D] fp32

    dim3 grid(N_TTILE, D / D_TILE, B);           // 16 x 8 x 32
    fsmn_fir_kernel<<<grid, 256, 0, stream>>>(x, filt, out);
}